// MAN_89713276879474
// MI455X (gfx1250) — compile-verified
//
#include <hip/hip_runtime.h>
#include <hip/hip_bf16.h>
#include <math.h>

typedef __attribute__((ext_vector_type(2))) float v2f;
typedef __attribute__((ext_vector_type(8))) float v8f;

#define B_DIM   16384
#define IN_DIM  512
#define H_DIM   512
#define MEM_DIM 2048

// float offsets inside d_out (concatenated outputs, return order)
#define OFF_M   (16384 * 1024)           // new_M      [2048, 512]
#define OFF_WU  (OFF_M + 2048 * 512)     // new_w_u    [16384, 2048] (temp: inner)
#define OFF_PWR (OFF_WU + 16384 * 2048)  // new_prev_w_r = w_r [16384, 2048]

#define GATE_C  0.401312339887548f
#define OMG_C   0.598687660112452f       // 1 - GATE
#define GAMMA_C 0.3f

// ---------------- block reduction helpers (blockDim.x == 256) ----------------
static __device__ __forceinline__ float blockReduceSum(float v, float* s) {
    int tid = threadIdx.x;
    s[tid] = v; __syncthreads();
    for (int k = 128; k > 0; k >>= 1) {
        if (tid < k) s[tid] += s[tid + k];
        __syncthreads();
    }
    float r = s[0]; __syncthreads();
    return r;
}

static __device__ __forceinline__ float blockReduceMax(float v, float* s) {
    int tid = threadIdx.x;
    s[tid] = v; __syncthreads();
    for (int k = 128; k > 0; k >>= 1) {
        if (tid < k) s[tid] = fmaxf(s[tid], s[tid + k]);
        __syncthreads();
    }
    float r = s[0]; __syncthreads();
    return r;
}

static __device__ __forceinline__ int blockArgMin(float v, int idx, float* s, int* si) {
    int tid = threadIdx.x;
    s[tid] = v; si[tid] = idx; __syncthreads();
    for (int k = 128; k > 0; k >>= 1) {
        if (tid < k) {
            float ov = s[tid + k]; int oi = si[tid + k];
            if (ov < s[tid] || (ov == s[tid] && oi < si[tid])) { s[tid] = ov; si[tid] = oi; }
        }
        __syncthreads();
    }
    int r = si[0]; __syncthreads();
    return r;
}

// ---------------- K0: per-row norms of M -> ws ----------------
__global__ void __launch_bounds__(256) k_mnorm(const float* __restrict__ M, float* __restrict__ mn) {
    __shared__ float sred[256];
    int m = blockIdx.x, tid = threadIdx.x;
    float a = M[(size_t)m * H_DIM + tid];
    float b = M[(size_t)m * H_DIM + tid + 256];
    float nrm = sqrtf(blockReduceSum(a * a + b * b, sred));
    if (tid == 0) mn[m] = nrm;
}

// ---------------- Kz: zero the k_w accumulator region ----------------
__global__ void __launch_bounds__(256) k_zero(float* __restrict__ p, int n) {
    for (int i = blockIdx.x * blockDim.x + threadIdx.x; i < n; i += gridDim.x * blockDim.x)
        p[i] = 0.0f;
}

// ---------------- K1: h = leaky_relu(X@W + b) -> out[:, 0:512] ----------------
// wave computes 64(M) x 16(N); f32 WMMA 16x16x4
__global__ void __launch_bounds__(256) k_gemm_h(const float* __restrict__ X, const float* __restrict__ W,
                                                const float* __restrict__ bias, float* __restrict__ out) {
    const int wave = blockIdx.x * 8 + (threadIdx.x >> 5);
    const int lane = threadIdx.x & 31;
    const int lm = lane & 15, hi = lane >> 4;
    const int nt = wave & 31;        // 512/16 = 32 N tiles
    const int mt = wave >> 5;        // 16384/64 = 256 M strips
    const int m0 = mt * 64, n0 = nt * 16;

    v8f acc[4];
    #pragma unroll
    for (int t = 0; t < 4; t++)
        #pragma unroll
        for (int r = 0; r < 8; r++) acc[t][r] = 0.0f;

    for (int k0 = 0; k0 < IN_DIM; k0 += 4) {
        const int kk = k0 + hi * 2;
        v2f a0 = *(const v2f*)(X + (size_t)(m0 + lm)      * IN_DIM + kk);
        v2f a1 = *(const v2f*)(X + (size_t)(m0 + 16 + lm) * IN_DIM + kk);
        v2f a2 = *(const v2f*)(X + (size_t)(m0 + 32 + lm) * IN_DIM + kk);
        v2f a3 = *(const v2f*)(X + (size_t)(m0 + 48 + lm) * IN_DIM + kk);
        v2f bv;
        bv[0] = W[(size_t)kk * H_DIM + n0 + lm];
        bv[1] = W[(size_t)(kk + 1) * H_DIM + n0 + lm];
        acc[0] = __builtin_amdgcn_wmma_f32_16x16x4_f32(false, a0, false, bv, (short)0, acc[0], false, false);
        acc[1] = __builtin_amdgcn_wmma_f32_16x16x4_f32(false, a1, false, bv, (short)0, acc[1], false, false);
        acc[2] = __builtin_amdgcn_wmma_f32_16x16x4_f32(false, a2, false, bv, (short)0, acc[2], false, false);
        acc[3] = __builtin_amdgcn_wmma_f32_16x16x4_f32(false, a3, false, bv, (short)0, acc[3], false, false);
    }
    float bn = bias[n0 + lm];
    #pragma unroll
    for (int t = 0; t < 4; t++)
        #pragma unroll
        for (int r = 0; r < 8; r++) {
            int m = m0 + t * 16 + r + hi * 8;
            float v = acc[t][r] + bn;
            v = v > 0.0f ? v : 0.01f * v;           // leaky_relu(0.01)
            out[(size_t)m * 1024 + n0 + lm] = v;
        }
}

// ---------------- K3: inner = h @ M^T -> out[OFF_WU] (temp) ----------------
__global__ void __launch_bounds__(256) k_gemm_inner(const float* __restrict__ M, float* __restrict__ out) {
    const int wave = blockIdx.x * 8 + (threadIdx.x >> 5);
    const int lane = threadIdx.x & 31;
    const int lm = lane & 15, hi = lane >> 4;
    const int nt = wave & 127;       // 2048/16 = 128 N tiles
    const int mt = wave >> 7;        // 256 M strips
    const int m0 = mt * 64, n0 = nt * 16;

    v8f acc[4];
    #pragma unroll
    for (int t = 0; t < 4; t++)
        #pragma unroll
        for (int r = 0; r < 8; r++) acc[t][r] = 0.0f;

    for (int k0 = 0; k0 < H_DIM; k0 += 4) {
        const int kk = k0 + hi * 2;
        v2f a0 = *(const v2f*)(out + (size_t)(m0 + lm)      * 1024 + kk);  // h rows, ld=1024
        v2f a1 = *(const v2f*)(out + (size_t)(m0 + 16 + lm) * 1024 + kk);
        v2f a2 = *(const v2f*)(out + (size_t)(m0 + 32 + lm) * 1024 + kk);
        v2f a3 = *(const v2f*)(out + (size_t)(m0 + 48 + lm) * 1024 + kk);
        // B(k,n) = M[n][k]  -> contiguous read of M row n
        v2f bv = *(const v2f*)(M + (size_t)(n0 + lm) * H_DIM + kk);
        acc[0] = __builtin_amdgcn_wmma_f32_16x16x4_f32(false, a0, false, bv, (short)0, acc[0], false, false);
        acc[1] = __builtin_amdgcn_wmma_f32_16x16x4_f32(false, a1, false, bv, (short)0, acc[1], false, false);
        acc[2] = __builtin_amdgcn_wmma_f32_16x16x4_f32(false, a2, false, bv, (short)0, acc[2], false, false);
        acc[3] = __builtin_amdgcn_wmma_f32_16x16x4_f32(false, a3, false, bv, (short)0, acc[3], false, false);
    }
    #pragma unroll
    for (int t = 0; t < 4; t++)
        #pragma unroll
        for (int r = 0; r < 8; r++) {
            int m = m0 + t * 16 + r + hi * 8;
            out[(size_t)OFF_WU + (size_t)m * MEM_DIM + n0 + lm] = acc[t][r];
        }
}

// ---------------- K4: fused per-batch-row ops ----------------
// softmax(w_r), argmin(w_u), new_w_u l2norm; writes w_r -> OFF_PWR, new_w_u -> OFF_WU (in place)
__global__ void __launch_bounds__(256) k_rows(const float* __restrict__ w_u, const float* __restrict__ prev_w_r,
                                              const float* __restrict__ mn, float* __restrict__ out,
                                              int* __restrict__ w_lu) {
    __shared__ float sred[256];
    __shared__ int   sidx[256];
    __shared__ float sv[MEM_DIM];
    const int b = blockIdx.x, tid = threadIdx.x;

    // h row norm
    const float* hrow = out + (size_t)b * 1024;
    float ss = 0.0f;
    for (int i = tid; i < H_DIM; i += 256) { float v = hrow[i]; ss += v * v; }
    float hn = sqrtf(blockReduceSum(ss, sred));

    // cosine logits + softmax
    float* inner_row = out + (size_t)OFF_WU + (size_t)b * MEM_DIM;
    float lmax = -3.402823466e38f;
    for (int j = tid; j < MEM_DIM; j += 256) {
        float l = inner_row[j] / (hn * mn[j] + 1e-8f);
        sv[j] = l;
        lmax = fmaxf(lmax, l);
    }
    float gmax = blockReduceMax(lmax, sred);
    float es = 0.0f;
    for (int j = tid; j < MEM_DIM; j += 256) { float e = expf(sv[j] - gmax); sv[j] = e; es += e; }
    float inv = 1.0f / blockReduceSum(es, sred);
    float* wr_row = out + (size_t)OFF_PWR + (size_t)b * MEM_DIM;
    for (int j = tid; j < MEM_DIM; j += 256) { float w = sv[j] * inv; sv[j] = w; wr_row[j] = w; }

    // argmin over w_u (first occurrence)
    const float* wu_row = w_u + (size_t)b * MEM_DIM;
    float mv = 3.402823466e38f; int mi = 0;
    for (int j = tid; j < MEM_DIM; j += 256) { float v = wu_row[j]; if (v < mv) { mv = v; mi = j; } }
    int lu = blockArgMin(mv, mi, sred, sidx);
    if (tid == 0) w_lu[b] = lu;

    // new_w_u = l2norm(GAMMA*w_u + w_r + w_w)
    const float* pwr_row = prev_w_r + (size_t)b * MEM_DIM;
    float ts = 0.0f;
    for (int j = tid; j < MEM_DIM; j += 256) {
        float ww = GATE_C * pwr_row[j] + ((j == lu) ? OMG_C : 0.0f);
        float t = GAMMA_C * wu_row[j] + sv[j] + ww;
        sv[j] = t; ts += t * t;
    }
    float inv2 = 1.0f / fmaxf(sqrtf(blockReduceSum(ts, sred)), 1e-12f);
    for (int j = tid; j < MEM_DIM; j += 256) inner_row[j] = sv[j] * inv2;   // overwrite temp with new_w_u
}

// ---------------- K5: read = w_r @ M -> out[:, 512:1024] ----------------
__global__ void __launch_bounds__(256) k_gemm_read(const float* __restrict__ M, float* __restrict__ out) {
    const int wave = blockIdx.x * 8 + (threadIdx.x >> 5);
    const int lane = threadIdx.x & 31;
    const int lm = lane & 15, hi = lane >> 4;
    const int nt = wave & 31;        // 32 N tiles
    const int mt = wave >> 5;        // 256 M strips
    const int m0 = mt * 64, n0 = nt * 16;
    const float* wr = out + OFF_PWR;

    v8f acc[4];
    #pragma unroll
    for (int t = 0; t < 4; t++)
        #pragma unroll
        for (int r = 0; r < 8; r++) acc[t][r] = 0.0f;

    for (int k0 = 0; k0 < MEM_DIM; k0 += 4) {
        const int kk = k0 + hi * 2;
        v2f a0 = *(const v2f*)(wr + (size_t)(m0 + lm)      * MEM_DIM + kk);
        v2f a1 = *(const v2f*)(wr + (size_t)(m0 + 16 + lm) * MEM_DIM + kk);
        v2f a2 = *(const v2f*)(wr + (size_t)(m0 + 32 + lm) * MEM_DIM + kk);
        v2f a3 = *(const v2f*)(wr + (size_t)(m0 + 48 + lm) * MEM_DIM + kk);
        v2f bv;
        bv[0] = M[(size_t)kk * H_DIM + n0 + lm];
        bv[1] = M[(size_t)(kk + 1) * H_DIM + n0 + lm];
        acc[0] = __builtin_amdgcn_wmma_f32_16x16x4_f32(false, a0, false, bv, (short)0, acc[0], false, false);
        acc[1] = __builtin_amdgcn_wmma_f32_16x16x4_f32(false, a1, false, bv, (short)0, acc[1], false, false);
        acc[2] = __builtin_amdgcn_wmma_f32_16x16x4_f32(false, a2, false, bv, (short)0, acc[2], false, false);
        acc[3] = __builtin_amdgcn_wmma_f32_16x16x4_f32(false, a3, false, bv, (short)0, acc[3], false, false);
    }
    #pragma unroll
    for (int t = 0; t < 4; t++)
        #pragma unroll
        for (int r = 0; r < 8; r++) {
            int m = m0 + t * 16 + r + hi * 8;
            out[(size_t)m * 1024 + 512 + n0 + lm] = acc[t][r];
        }
}

// ---------------- K6: k_w = w_w^T @ h, K split 16 ways, atomic accumulate ----------------
__global__ void __launch_bounds__(256) k_gemm_kw(const float* __restrict__ prev_w_r,
                                                 const int* __restrict__ w_lu, float* __restrict__ out) {
    const int wave = blockIdx.x * 8 + (threadIdx.x >> 5);   // 16384 waves
    const int lane = threadIdx.x & 31;
    const int lm = lane & 15, hi = lane >> 4;
    const int ks = wave >> 10;                              // 16 K splits
    const int t0 = wave & 1023;
    const int mt = t0 >> 5;                                 // 2048/64 = 32 M strips (mem slots)
    const int nt = t0 & 31;                                 // 512/16 = 32 N tiles (H)
    const int m0 = mt * 64, n0 = nt * 16;

    v8f acc[4];
    #pragma unroll
    for (int t = 0; t < 4; t++)
        #pragma unroll
        for (int r = 0; r < 8; r++) acc[t][r] = 0.0f;

    const int kbeg = ks * 1024, kend = kbeg + 1024;
    for (int k0 = kbeg; k0 < kend; k0 += 4) {
        const int kk = k0 + hi * 2;
        __builtin_prefetch(prev_w_r + (size_t)(kk + 32) * MEM_DIM + m0 + lm, 0, 0);
        const int lu0 = w_lu[kk], lu1 = w_lu[kk + 1];
        v2f a[4];
        #pragma unroll
        for (int t = 0; t < 4; t++) {
            const int m = m0 + t * 16 + lm;
            a[t][0] = GATE_C * prev_w_r[(size_t)kk * MEM_DIM + m]       + ((lu0 == m) ? OMG_C : 0.0f);
            a[t][1] = GATE_C * prev_w_r[(size_t)(kk + 1) * MEM_DIM + m] + ((lu1 == m) ? OMG_C : 0.0f);
        }
        v2f bv;                                              // B(k,n) = h[k][n]
        bv[0] = out[(size_t)kk * 1024 + n0 + lm];
        bv[1] = out[(size_t)(kk + 1) * 1024 + n0 + lm];
        acc[0] = __builtin_amdgcn_wmma_f32_16x16x4_f32(false, a[0], false, bv, (short)0, acc[0], false, false);
        acc[1] = __builtin_amdgcn_wmma_f32_16x16x4_f32(false, a[1], false, bv, (short)0, acc[1], false, false);
        acc[2] = __builtin_amdgcn_wmma_f32_16x16x4_f32(false, a[2], false, bv, (short)0, acc[2], false, false);
        acc[3] = __builtin_amdgcn_wmma_f32_16x16x4_f32(false, a[3], false, bv, (short)0, acc[3], false, false);
    }
    #pragma unroll
    for (int t = 0; t < 4; t++)
        #pragma unroll
        for (int r = 0; r < 8; r++) {
            int m = m0 + t * 16 + r + hi * 8;
            atomicAdd(out + (size_t)OFF_M + (size_t)m * H_DIM + n0 + lm, acc[t][r]);
        }
}

// ---------------- K7: new_M = l2norm(M + k_w), in place over OFF_M ----------------
__global__ void __launch_bounds__(256) k_newM(const float* __restrict__ M, float* __restrict__ out) {
    __shared__ float sred[256];
    const int m = blockIdx.x, tid = threadIdx.x;
    float* row = out + (size_t)OFF_M + (size_t)m * H_DIM;
    const float* Mr = M + (size_t)m * H_DIM;
    float v0 = Mr[tid] + row[tid];
    float v1 = Mr[tid + 256] + row[tid + 256];
    float inv = 1.0f / fmaxf(sqrtf(blockReduceSum(v0 * v0 + v1 * v1, sred)), 1e-12f);
    row[tid] = v0 * inv;
    row[tid + 256] = v1 * inv;
}

extern "C" void kernel_launch(void* const* d_in, const int* in_sizes, int n_in,
                              void* d_out, int out_size, void* d_ws, size_t ws_size,
                              hipStream_t stream) {
    const float* X    = (const float*)d_in[0];
    const float* W    = (const float*)d_in[1];
    const float* bias = (const float*)d_in[2];
    const float* M    = (const float*)d_in[3];
    const float* w_u  = (const float*)d_in[4];
    const float* pwr  = (const float*)d_in[5];
    float* out = (float*)d_out;
    float* mn  = (float*)d_ws;                                  // [2048]
    int*   wlu = (int*)((char*)d_ws + MEM_DIM * sizeof(float)); // [16384]

    k_mnorm     <<<MEM_DIM, 256, 0, stream>>>(M, mn);
    k_zero      <<<1024,    256, 0, stream>>>(out + OFF_M, MEM_DIM * H_DIM);
    k_gemm_h    <<<1024,    256, 0, stream>>>(X, W, bias, out);       // 8192 waves
    k_gemm_inner<<<4096,    256, 0, stream>>>(M, out);                // 32768 waves
    k_rows      <<<B_DIM,   256, 0, stream>>>(w_u, pwr, mn, out, wlu);
    k_gemm_read <<<1024,    256, 0, stream>>>(M, out);                // 8192 waves
    k_gemm_kw   <<<2048,    256, 0, stream>>>(pwr, wlu, out);         // 16384 waves
    k_newM      <<<MEM_DIM, 256, 0, stream>>>(M, out);
}